// GAT_69441031242185
// MI455X (gfx1250) — compile-verified
//
#include <hip/hip_runtime.h>
#include <math.h>

// ---------------- problem constants (from reference) ----------------
#define NN_NODES 20000
#define NE       320000
#define ETOT     (NE + NN_NODES)   // edges + self loops = 340000
#define F_IN     128
#define CC       64
#define HH       8
#define HC       (HH * CC)         // 512
#define NEG_SLOPE 0.2f
#define KP       128               // K-panel rows staged in LDS (32KB)

typedef __attribute__((ext_vector_type(2))) float v2f;
typedef __attribute__((ext_vector_type(8))) float v8f;

// ---------------- utility kernels ----------------
__global__ void fill_f32(float* p, long n, float v) {
    long i = (long)blockIdx.x * blockDim.x + threadIdx.x;
    if (i < n) p[i] = v;
}

__device__ inline void atomicMaxF(float* addr, float v) {
    // works for all finite values when addr initialized to -inf
    if (v >= 0.0f) atomicMax((int*)addr, __float_as_int(v));
    else           atomicMin((unsigned int*)addr, __float_as_uint(v));
}

// ---------------- WMMA f32 GEMM: C[M,NN] = A[M,K] * B[K,NN] ----------------
// Block = 256 threads = 8 waves, all sharing one 64-column strip (col0).
// Wave w computes M-tile (mblk*8+w). B K-panel (KP x 64) is staged into LDS
// once per block via async global->LDS loads, then fed to
// V_WMMA_F32_16X16X4_F32 through conflict-free ds_load_b64.
// Requirements: M%16==0, NN%64==0, K%KP==0.
__global__ void gemm_f32_wmma(const float* __restrict__ A,
                              const float* __restrict__ B,
                              float* __restrict__ C,
                              int M, int K, int NNc) {
    // LDS layout (float2 units): idx2 = khalf*64 + (col ^ ((khalf&1)<<4))
    // where khalf = k>>1 ; the float2 holds rows (2*khalf, 2*khalf+1) at col.
    __shared__ float2 ldsB2[(KP / 2) * 64];          // 32 KB

    int tid  = threadIdx.x;
    int lane = tid & 31;
    int w    = tid >> 5;                              // wave in block: 0..7
    int tilesN64 = NNc >> 6;
    int mtiles   = M >> 4;
    int tg   = blockIdx.x % tilesN64;
    int mblk = blockIdx.x / tilesN64;
    int tm   = mblk * 8 + w;
    bool valid = (tm < mtiles);

    int row0   = tm << 4;
    int col0   = tg << 6;
    int laneLo = lane & 15;
    int hs     = lane >> 4;                           // half-wave K select

    const float* Ap = A + (size_t)(row0 + laneLo) * K + 2 * hs;
    unsigned ldsBase = (unsigned)(uintptr_t)(&ldsB2[0]);   // LDS byte address

    // hoisted swizzled column indices for ds reads (float2 units)
    int x0 = (laneLo +  0) ^ (hs << 4);
    int x1 = (laneLo + 16) ^ (hs << 4);
    int x2 = (laneLo + 32) ^ (hs << 4);
    int x3 = (laneLo + 48) ^ (hs << 4);

    v8f acc0 = {}; v8f acc1 = {}; v8f acc2 = {}; v8f acc3 = {};

    for (int kp0 = 0; kp0 < K; kp0 += KP) {
        if (kp0) __syncthreads();                     // protect prior reads

        // ---- cooperative async fill of B panel into LDS (swizzled) ----
        for (int idx = tid; idx < KP * 64; idx += 256) {
            int kk = idx >> 6;                        // panel row 0..KP-1
            int c  = idx & 63;                        // column within strip
            int khalf = kk >> 1;
            unsigned f2i    = (unsigned)(khalf * 64 + (c ^ ((khalf & 1) << 4)));
            unsigned ldsOff = ldsBase + f2i * 8u + (unsigned)(kk & 1) * 4u;
            const float* gp = B + (size_t)(kp0 + kk) * NNc + col0 + c;
            asm volatile("global_load_async_to_lds_b32 %0, %1, off"
                         :: "v"(ldsOff), "v"(gp) : "memory");
        }
        asm volatile("s_wait_asynccnt 0x0" ::: "memory");
        __syncthreads();

        // ---- compute over this K panel ----
        if (valid) {
            #pragma unroll 4
            for (int k = 0; k < KP; k += 4) {
                __builtin_prefetch(Ap + kp0 + k + 32, 0, 0);
                float2 av = *(const float2*)(Ap + kp0 + k);
                v2f a; a.x = av.x; a.y = av.y;

                const float2* lb = ldsB2 + ((k >> 1) + hs) * 64;
                float2 f0 = lb[x0], f1 = lb[x1], f2 = lb[x2], f3 = lb[x3];
                v2f b0; b0.x = f0.x; b0.y = f0.y;
                v2f b1; b1.x = f1.x; b1.y = f1.y;
                v2f b2; b2.x = f2.x; b2.y = f2.y;
                v2f b3; b3.x = f3.x; b3.y = f3.y;

                acc0 = __builtin_amdgcn_wmma_f32_16x16x4_f32(false, a, false, b0, (short)0, acc0, false, false);
                acc1 = __builtin_amdgcn_wmma_f32_16x16x4_f32(false, a, false, b1, (short)0, acc1, false, false);
                acc2 = __builtin_amdgcn_wmma_f32_16x16x4_f32(false, a, false, b2, (short)0, acc2, false, false);
                acc3 = __builtin_amdgcn_wmma_f32_16x16x4_f32(false, a, false, b3, (short)0, acc3, false, false);
            }
        }
    }

    if (!valid) return;
    // C/D layout: VGPR r -> row r + 8*hs, col = laneLo (+16 per tile)
    #pragma unroll
    for (int r = 0; r < 8; ++r) {
        size_t row = (size_t)(row0 + (hs << 3) + r);
        float* Cp = C + row * NNc + col0 + laneLo;
        Cp[0]  = acc0[r];
        Cp[16] = acc1[r];
        Cp[32] = acc2[r];
        Cp[48] = acc3[r];
    }
}

// ---------------- attention logit projections ----------------
template<int H>
__global__ void al_kernel(const float* __restrict__ hmat,
                          const float* __restrict__ asrc,
                          const float* __restrict__ adst,
                          float* __restrict__ os, float* __restrict__ od) {
    int idx = blockIdx.x * blockDim.x + threadIdx.x;
    if (idx >= NN_NODES * H) return;
    int n = idx / H, h = idx % H;
    const float4* hp = (const float4*)(hmat + (size_t)n * (H * CC) + h * CC);
    const float4* as = (const float4*)(asrc + h * CC);
    const float4* ad = (const float4*)(adst + h * CC);
    float ss = 0.f, sd = 0.f;
    #pragma unroll
    for (int j = 0; j < CC / 4; ++j) {
        float4 v = hp[j], a = as[j], b = ad[j];
        ss += v.x * a.x + v.y * a.y + v.z * a.z + v.w * a.w;
        sd += v.x * b.x + v.y * b.y + v.z * b.z + v.w * b.w;
    }
    os[idx] = ss;
    od[idx] = sd;
}

__device__ inline void edge_endpoints(int e, const int* ei, int& src, int& dst) {
    if (e < NE) { src = ei[e]; dst = ei[NE + e]; }
    else        { src = e - NE; dst = e - NE; }          // self loop
}

// ---------------- edge pass 1: segment max ----------------
template<int H>
__global__ void edge_max_k(const int* __restrict__ ei,
                           const float* __restrict__ als,
                           const float* __restrict__ ald,
                           float* __restrict__ m) {
    int idx = blockIdx.x * blockDim.x + threadIdx.x;
    if (idx >= ETOT * H) return;
    int e = idx / H, h = idx % H;
    int src, dst; edge_endpoints(e, ei, src, dst);
    float v = als[src * H + h] + ald[dst * H + h];
    v = v > 0.f ? v : NEG_SLOPE * v;
    atomicMaxF(&m[dst * H + h], v);
}

// ---------------- edge pass 2: exp + segment sum ----------------
template<int H>
__global__ void edge_exp_k(const int* __restrict__ ei,
                           const float* __restrict__ als,
                           const float* __restrict__ ald,
                           const float* __restrict__ m,
                           float* __restrict__ z,
                           float* __restrict__ expe) {
    int idx = blockIdx.x * blockDim.x + threadIdx.x;
    if (idx >= ETOT * H) return;
    int e = idx / H, h = idx % H;
    int src, dst; edge_endpoints(e, ei, src, dst);
    float v = als[src * H + h] + ald[dst * H + h];
    v = v > 0.f ? v : NEG_SLOPE * v;
    float ex = expf(v - m[dst * H + h]);
    expe[idx] = ex;                                   // idx == e*H + h
    atomicAdd(&z[dst * H + h], ex);
}

// ---------------- edge pass 3 (layer 0): aggregate, wave per edge ----------------
__global__ void edge_agg0_k(const int* __restrict__ ei,
                            const float* __restrict__ h0,
                            const float* __restrict__ expe,
                            const float* __restrict__ z,
                            float* __restrict__ out) {
    int gid  = blockIdx.x * blockDim.x + threadIdx.x;
    int wid  = gid >> 5;
    int lane = gid & 31;
    if (wid >= ETOT) return;
    int src, dst; edge_endpoints(wid, ei, src, dst);
    const float4* hp = (const float4*)(h0 + (size_t)src * HC);
    #pragma unroll
    for (int j = 0; j < 4; ++j) {
        int c = lane * 4 + j * 128;                   // 4 channels, same head
        int h = c >> 6;
        float alpha = expe[(size_t)wid * HH + h] / (z[(size_t)dst * HH + h] + 1e-16f);
        float4 v = hp[lane + j * 32];
        float* op = out + (size_t)dst * HC + c;
        atomicAdd(op + 0, v.x * alpha);
        atomicAdd(op + 1, v.y * alpha);
        atomicAdd(op + 2, v.z * alpha);
        atomicAdd(op + 3, v.w * alpha);
    }
}

// ---------------- edge pass 3 (layer 1): thread per (edge, channel) ----------------
__global__ void edge_agg1_k(const int* __restrict__ ei,
                            const float* __restrict__ h1,
                            const float* __restrict__ expe,
                            const float* __restrict__ z,
                            float* __restrict__ out) {
    long gid = (long)blockIdx.x * blockDim.x + threadIdx.x;
    if (gid >= (long)ETOT * CC) return;
    int e = (int)(gid >> 6);
    int c = (int)(gid & 63);
    int src, dst; edge_endpoints(e, ei, src, dst);
    float alpha = expe[e] / (z[dst] + 1e-16f);
    atomicAdd(&out[(size_t)dst * CC + c], h1[(size_t)src * CC + c] * alpha);
}

// ---------------- bias + ELU in place over out0 [N,512] ----------------
__global__ void bias_elu_k(float* __restrict__ out0, const float* __restrict__ b0) {
    long i = (long)blockIdx.x * blockDim.x + threadIdx.x;
    if (i >= (long)NN_NODES * HC) return;
    float x = out0[i] + b0[i % HC];
    out0[i] = x > 0.f ? x : (expf(x) - 1.f);
}

// ---------------- mean pool (partial sums) ----------------
__global__ void pool_partial_k(const float* __restrict__ out1, float* __restrict__ pool) {
    int c    = threadIdx.x & 63;
    int rsub = threadIdx.x >> 6;                       // 0..3
    int r0   = blockIdx.x * 80;
    float s = 0.f;
    for (int r = r0 + rsub; r < r0 + 80 && r < NN_NODES; r += 4)
        s += out1[(size_t)r * CC + c];
    atomicAdd(&pool[c], s);
}

// ---------------- classifier + sigmoid (1 wave) ----------------
__global__ void finalize_k(const float* __restrict__ pool,
                           const float* __restrict__ b1,
                           const float* __restrict__ Wc,
                           const float* __restrict__ bc,
                           float* __restrict__ out) {
    int l = threadIdx.x;                               // 32 lanes
    float s = 0.f;
    for (int c = l; c < CC; c += 32) {
        float p = pool[c] * (1.0f / NN_NODES) + b1[c]; // mean(out1)+b1
        s += p * Wc[c];
    }
    for (int off = 16; off; off >>= 1) s += __shfl_down(s, off, 32);
    if (l == 0) out[0] = 1.f / (1.f + expf(-(s + bc[0])));
}

// ---------------- driver ----------------
static inline int cdiv(long a, int b) { return (int)((a + b - 1) / b); }

extern "C" void kernel_launch(void* const* d_in, const int* in_sizes, int n_in,
                              void* d_out, int out_size, void* d_ws, size_t ws_size,
                              hipStream_t stream) {
    const float* x      = (const float*)d_in[0];
    const int*   ei     = (const int*)  d_in[1];   // int32 (jax x64 off)
    const float* W0     = (const float*)d_in[2];
    const float* a_src0 = (const float*)d_in[3];
    const float* a_dst0 = (const float*)d_in[4];
    const float* b0     = (const float*)d_in[5];
    const float* W1     = (const float*)d_in[6];
    const float* a_src1 = (const float*)d_in[7];
    const float* a_dst1 = (const float*)d_in[8];
    const float* b1     = (const float*)d_in[9];
    const float* Wc     = (const float*)d_in[10];
    const float* bc     = (const float*)d_in[11];
    float* outp = (float*)d_out;

    // workspace carve (floats)
    float* ws = (float*)d_ws;
    size_t o = 0;
    float* h0    = ws + o; o += (size_t)NN_NODES * HC;   // [N,512]
    float* out0  = ws + o; o += (size_t)NN_NODES * HC;   // [N,512] agg / layer1 input
    float* als0  = ws + o; o += (size_t)NN_NODES * HH;
    float* ald0  = ws + o; o += (size_t)NN_NODES * HH;
    float* m0    = ws + o; o += (size_t)NN_NODES * HH;
    float* z0    = ws + o; o += (size_t)NN_NODES * HH;
    float* expe0 = ws + o; o += (size_t)ETOT * HH;
    float* h1    = ws + o; o += (size_t)NN_NODES * CC;
    float* out1  = ws + o; o += (size_t)NN_NODES * CC;
    float* als1  = ws + o; o += (size_t)NN_NODES;
    float* ald1  = ws + o; o += (size_t)NN_NODES;
    float* m1    = ws + o; o += (size_t)NN_NODES;
    float* z1    = ws + o; o += (size_t)NN_NODES;
    float* expe1 = ws + o; o += (size_t)ETOT;
    float* pool  = ws + o; o += 64;

    const int B = 256;
    const float NEG_INF = -INFINITY;

    // ---- init accumulators (every call; harness never re-poisons) ----
    fill_f32<<<cdiv((long)NN_NODES * HC, B), B, 0, stream>>>(out0, (long)NN_NODES * HC, 0.f);
    fill_f32<<<cdiv((long)NN_NODES * HH, B), B, 0, stream>>>(m0, (long)NN_NODES * HH, NEG_INF);
    fill_f32<<<cdiv((long)NN_NODES * HH, B), B, 0, stream>>>(z0, (long)NN_NODES * HH, 0.f);
    fill_f32<<<cdiv((long)NN_NODES * CC, B), B, 0, stream>>>(out1, (long)NN_NODES * CC, 0.f);
    fill_f32<<<cdiv(NN_NODES, B), B, 0, stream>>>(m1, NN_NODES, NEG_INF);
    fill_f32<<<cdiv(NN_NODES, B), B, 0, stream>>>(z1, NN_NODES, 0.f);
    fill_f32<<<1, 64, 0, stream>>>(pool, 64, 0.f);

    // ---- layer 0 ----
    {   // h0 = x @ W0   (20000x128 @ 128x512)
        int mblks = cdiv(NN_NODES / 16, 8);
        int blocks = mblks * (HC / 64);
        gemm_f32_wmma<<<blocks, B, 0, stream>>>(x, W0, h0, NN_NODES, F_IN, HC);
    }
    al_kernel<HH><<<cdiv((long)NN_NODES * HH, B), B, 0, stream>>>(h0, a_src0, a_dst0, als0, ald0);
    edge_max_k<HH><<<cdiv((long)ETOT * HH, B), B, 0, stream>>>(ei, als0, ald0, m0);
    edge_exp_k<HH><<<cdiv((long)ETOT * HH, B), B, 0, stream>>>(ei, als0, ald0, m0, z0, expe0);
    edge_agg0_k<<<cdiv((long)ETOT * 32, B), B, 0, stream>>>(ei, h0, expe0, z0, out0);
    bias_elu_k<<<cdiv((long)NN_NODES * HC, B), B, 0, stream>>>(out0, b0);

    // ---- layer 1 ----
    {   // h1 = elu(out0) @ W1   (20000x512 @ 512x64)
        int mblks = cdiv(NN_NODES / 16, 8);
        int blocks = mblks * (CC / 64);
        gemm_f32_wmma<<<blocks, B, 0, stream>>>(out0, W1, h1, NN_NODES, HC, CC);
    }
    al_kernel<1><<<cdiv(NN_NODES, B), B, 0, stream>>>(h1, a_src1, a_dst1, als1, ald1);
    edge_max_k<1><<<cdiv(ETOT, B), B, 0, stream>>>(ei, als1, ald1, m1);
    edge_exp_k<1><<<cdiv(ETOT, B), B, 0, stream>>>(ei, als1, ald1, m1, z1, expe1);
    edge_agg1_k<<<cdiv((long)ETOT * CC, B), B, 0, stream>>>(ei, h1, expe1, z1, out1);

    // ---- pool + classifier ----
    pool_partial_k<<<cdiv(NN_NODES, 80), B, 0, stream>>>(out1, pool);
    finalize_k<<<1, 32, 0, stream>>>(pool, b1, Wc, bc, outp);
    (void)in_sizes; (void)n_in; (void)out_size; (void)ws_size;
}